// GCN_22832046146280
// MI455X (gfx1250) — compile-verified
//
#include <hip/hip_runtime.h>
#include <math.h>

typedef __attribute__((ext_vector_type(2))) float v2f;
typedef __attribute__((ext_vector_type(8))) float v8f;
typedef __attribute__((ext_vector_type(4))) int v4i;

// gfx1250 async global->LDS path (guarded: falls back to plain LDS staging)
#if __has_builtin(__builtin_amdgcn_global_load_async_to_lds_b128) && \
    __has_builtin(__builtin_amdgcn_s_wait_asynccnt)
#define USE_ASYNC_LDS 1
#else
#define USE_ASYNC_LDS 0
#endif

#if USE_ASYNC_LDS
// Probe-derived signature: param 0 is (AS1) int4*, param 1 is (AS3) int4*,
// then immediate offset + cpol.
__device__ __forceinline__ void async_copy16(const void* g, void* l) {
  __builtin_amdgcn_global_load_async_to_lds_b128(
      (__attribute__((address_space(1))) v4i*)(g),
      (__attribute__((address_space(3))) v4i*)(l),
      /*offset=*/0, /*cpol=*/0);
}
#endif

// ---------------------------------------------------------------------------
// fills
// ---------------------------------------------------------------------------
__global__ void fill_kernel(float* __restrict__ p, float v, long long n) {
  long long i = (long long)blockIdx.x * blockDim.x + threadIdx.x;
  if (i < n) p[i] = v;
}

__global__ void fill4_kernel(float4* __restrict__ p, float v, long long nq) {
  long long i = (long long)blockIdx.x * blockDim.x + threadIdx.x;
  if (i < nq) p[i] = make_float4(v, v, v, v);
}

// ---------------------------------------------------------------------------
// in-degree accumulation over edge destinations (deg pre-filled with 1.0 for
// the self-loop), then in-place rsqrt
// ---------------------------------------------------------------------------
__global__ void degree_kernel(const int* __restrict__ dst, float* __restrict__ deg, int E) {
  int e = blockIdx.x * blockDim.x + threadIdx.x;
  if (e < E) atomicAdd(&deg[dst[e]], 1.0f);
}

__global__ void rsqrt_kernel(float* __restrict__ deg, int n) {
  int i = blockIdx.x * blockDim.x + threadIdx.x;
  if (i < n) {
    float d = deg[i];
    deg[i] = d > 0.0f ? rsqrtf(d) : 0.0f;
  }
}

// ---------------------------------------------------------------------------
// fp32 WMMA GEMM: C[M x N] = A[M x K] @ B[K x N], N = NTILES*16, K % 4 == 0,
// M % 16 == 0. Block = 256 threads = 8 waves. B is staged once per block into
// LDS (async b128 copy when available), then each wave computes one 16-row
// strip across ALL N-tiles: A is loaded once per K-step and reused by NTILES
// V_WMMA_F32_16X16X4_F32 ops (4x less A traffic than tile-per-wave).
//
// Lane layouts per CDNA5 ISA 7.12.2:
//   A (16x4): lane m=L&15; VGPR0 = K = (L>=16 ? 2 : 0), VGPR1 = that + 1
//   B (4x16): col n=L&15;  VGPR0 = row K = (L>=16 ? 2 : 0), VGPR1 = that + 1
//   C/D (16x16): VGPR v holds row v (lanes 0-15) / row v+8 (lanes 16-31)
// ---------------------------------------------------------------------------
template <int K, int NTILES>
__global__ __launch_bounds__(256) void gemm_wmma_f32(const float* __restrict__ A,
                                                     const float* __restrict__ B,
                                                     float* __restrict__ C, int M) {
  constexpr int N = NTILES * 16;
  __shared__ float Bs[K * N];

  // ---- stage B (K*N*4 bytes) into LDS; all 256 threads cooperate ----
#if USE_ASYNC_LDS
  {
    constexpr int ITERS = (K * N * 4) / (256 * 16);  // b128 per lane per iter
    const int t = threadIdx.x;
#pragma unroll
    for (int it = 0; it < ITERS; ++it) {
      const int off = (it * 256 + t) * 16;
      async_copy16((const char*)B + off, (char*)Bs + off);
    }
    __builtin_amdgcn_s_wait_asynccnt(0);
  }
#else
  for (int i = threadIdx.x; i < K * N; i += 256) Bs[i] = B[i];
#endif
  __syncthreads();

  const int mt = blockIdx.x * 8 + (threadIdx.x >> 5);
  if (mt * 16 >= M) return;  // wave-uniform (after barrier): EXEC all-ones

  const int lane = threadIdx.x & 31;
  const int col = lane & 15;  // A-row within tile / B,C column within tile
  const int hs = lane >> 4;   // lane-half select

  const float* Arow = A + (long long)(mt * 16 + col) * K;

  v8f acc[NTILES];
  v8f zero = {};
#pragma unroll
  for (int n = 0; n < NTILES; ++n) acc[n] = zero;

#pragma unroll 4
  for (int k0 = 0; k0 < K; k0 += 4) {
    const int ka = k0 + hs * 2;
    v2f a = {Arow[ka], Arow[ka + 1]};
#pragma unroll
    for (int n = 0; n < NTILES; ++n) {
      v2f b = {Bs[ka * N + n * 16 + col], Bs[(ka + 1) * N + n * 16 + col]};
      acc[n] = __builtin_amdgcn_wmma_f32_16x16x4_f32(
          /*neg_a=*/false, a, /*neg_b=*/false, b,
          /*c_mod=*/(short)0, acc[n], /*reuse_a=*/false, /*reuse_b=*/false);
    }
  }

  const long long base = (long long)(mt * 16 + hs * 8) * N + col;
#pragma unroll
  for (int n = 0; n < NTILES; ++n)
#pragma unroll
    for (int v = 0; v < 8; ++v)
      C[base + n * 16 + (long long)v * N] = acc[n][v];
}

// ---------------------------------------------------------------------------
// edge scatter: agg[dst] += h[src] * dinv[src]*dinv[dst]; one thread per
// (edge, 4-feature quad): b128 gather + 4 f32 atomics. All intermediates are
// L2-resident (<< 192MB), so gathers + atomics stay on-die.
// ---------------------------------------------------------------------------
template <int LOGF>  // F = 1<<LOGF features, LOGF >= 2
__global__ void scatter_kernel(const float4* __restrict__ h,
                               const int* __restrict__ src,
                               const int* __restrict__ dst,
                               const float* __restrict__ dinv,
                               float* __restrict__ agg, long long total_quads) {
  constexpr int LOGQ = LOGF - 2;
  long long idx = (long long)blockIdx.x * blockDim.x + threadIdx.x;
  if (idx >= total_quads) return;
  long long e = idx >> LOGQ;
  int q = (int)(idx & ((1 << LOGQ) - 1));
  int s = src[e];
  int d = dst[e];
  float nrm = dinv[s] * dinv[d];
  float4 hv = h[((long long)s << LOGQ) + q];
  float* out = agg + ((long long)d << LOGF) + q * 4;
  atomicAdd(out + 0, hv.x * nrm);
  atomicAdd(out + 1, hv.y * nrm);
  atomicAdd(out + 2, hv.z * nrm);
  atomicAdd(out + 3, hv.w * nrm);
}

// ---------------------------------------------------------------------------
// layer-1 finalize (in place): agg = relu(agg + h*dinv^2 (self loop) + bias)
// ---------------------------------------------------------------------------
template <int LOGF>
__global__ void finalize_relu_kernel(float4* __restrict__ agg, const float4* __restrict__ h,
                                     const float* __restrict__ dinv,
                                     const float4* __restrict__ bias4, long long total_quads) {
  constexpr int LOGQ = LOGF - 2;
  long long idx = (long long)blockIdx.x * blockDim.x + threadIdx.x;
  if (idx >= total_quads) return;
  long long i = idx >> LOGQ;
  int q = (int)(idx & ((1 << LOGQ) - 1));
  float di = dinv[i];
  float di2 = di * di;
  float4 a = agg[idx];
  float4 hv = h[idx];
  float4 b = bias4[q];
  float4 r;
  r.x = fmaxf(a.x + hv.x * di2 + b.x, 0.0f);
  r.y = fmaxf(a.y + hv.y * di2 + b.y, 0.0f);
  r.z = fmaxf(a.z + hv.z * di2 + b.z, 0.0f);
  r.w = fmaxf(a.w + hv.w * di2 + b.w, 0.0f);
  agg[idx] = r;
}

// ---------------------------------------------------------------------------
// layer-2 finalize: add self loop + bias, then row-wise log_softmax (16 cols)
// ---------------------------------------------------------------------------
__global__ void finalize_logsoftmax_kernel(const float4* __restrict__ agg2,
                                           const float4* __restrict__ h2,
                                           const float* __restrict__ dinv,
                                           const float4* __restrict__ bias4,
                                           float4* __restrict__ out, int n) {
  int i = blockIdx.x * blockDim.x + threadIdx.x;
  if (i >= n) return;
  float di = dinv[i];
  float di2 = di * di;
  float vals[16];
  float mx = -1e30f;
#pragma unroll
  for (int qq = 0; qq < 4; ++qq) {
    float4 a = agg2[(long long)i * 4 + qq];
    float4 hv = h2[(long long)i * 4 + qq];
    float4 b = bias4[qq];
    vals[qq * 4 + 0] = a.x + hv.x * di2 + b.x;
    vals[qq * 4 + 1] = a.y + hv.y * di2 + b.y;
    vals[qq * 4 + 2] = a.z + hv.z * di2 + b.z;
    vals[qq * 4 + 3] = a.w + hv.w * di2 + b.w;
  }
#pragma unroll
  for (int f = 0; f < 16; ++f) mx = fmaxf(mx, vals[f]);
  float s = 0.0f;
#pragma unroll
  for (int f = 0; f < 16; ++f) s += expf(vals[f] - mx);
  float ls = mx + logf(s);
#pragma unroll
  for (int qq = 0; qq < 4; ++qq) {
    float4 r;
    r.x = vals[qq * 4 + 0] - ls;
    r.y = vals[qq * 4 + 1] - ls;
    r.z = vals[qq * 4 + 2] - ls;
    r.w = vals[qq * 4 + 3] - ls;
    out[(long long)i * 4 + qq] = r;
  }
}

// ---------------------------------------------------------------------------
// launch
// ---------------------------------------------------------------------------
extern "C" void kernel_launch(void* const* d_in, const int* in_sizes, int n_in,
                              void* d_out, int out_size, void* d_ws, size_t ws_size,
                              hipStream_t stream) {
  const int FIN = 128, FHID = 64, FOUT = 16;

  const float* x  = (const float*)d_in[0];
  const int*   ei = (const int*)d_in[1];  // [2, E] int32 (JAX default x64-off)
  const float* W1 = (const float*)d_in[2];
  const float* b1 = (const float*)d_in[3];
  const float* W2 = (const float*)d_in[4];
  const float* b2 = (const float*)d_in[5];

  const int N = in_sizes[0] / FIN;  // 100000 (multiple of 16)
  const int E = in_sizes[1] / 2;    // 3200000
  const int* src = ei;
  const int* dst = ei + E;

  // workspace (floats): dinv[N] | h1[N*64] | agg1[N*64] | h2[N*16] | agg2[N*16]
  float* ws   = (float*)d_ws;
  float* dinv = ws;
  float* h1   = dinv + N;
  float* agg1 = h1 + (long long)N * FHID;
  float* h2   = agg1 + (long long)N * FHID;
  float* agg2 = h2 + (long long)N * FOUT;

  const int B = 256;
  auto blocks = [](long long n, int b) { return (unsigned int)((n + b - 1) / b); };

  // 1) symmetric norm: deg = 1 (self loop) + in-degree; dinv = rsqrt(deg)
  fill_kernel<<<blocks(N, B), B, 0, stream>>>(dinv, 1.0f, N);
  degree_kernel<<<blocks(E, B), B, 0, stream>>>(dst, dinv, E);
  rsqrt_kernel<<<blocks(N, B), B, 0, stream>>>(dinv, N);

  // zero scatter accumulators (ws is poisoned by harness; re-zero every call)
  const long long q1 = (long long)N * FHID / 4;
  const long long q2 = (long long)N * FOUT / 4;
  fill4_kernel<<<blocks(q1, B), B, 0, stream>>>((float4*)agg1, 0.0f, q1);
  fill4_kernel<<<blocks(q2, B), B, 0, stream>>>((float4*)agg2, 0.0f, q2);

  // 2) h1 = x @ W1 : 6250 wave-strips of 16x64, 8 waves per block
  gemm_wmma_f32<128, 4><<<blocks((long long)(N / 16) * 32, B), B, 0, stream>>>(x, W1, h1, N);

  // 3) layer-1 edge aggregation + self-loop/bias/relu (agg1 -> relu'd h, in place)
  scatter_kernel<6><<<blocks((long long)E * (FHID / 4), B), B, 0, stream>>>(
      (const float4*)h1, src, dst, dinv, agg1, (long long)E * (FHID / 4));
  finalize_relu_kernel<6><<<blocks(q1, B), B, 0, stream>>>(
      (float4*)agg1, (const float4*)h1, dinv, (const float4*)b1, q1);

  // 4) h2 = relu(h) @ W2 : 6250 wave-strips of 16x16
  gemm_wmma_f32<64, 1><<<blocks((long long)(N / 16) * 32, B), B, 0, stream>>>(agg1, W2, h2, N);

  // 5) layer-2 edge aggregation + self-loop/bias + log_softmax
  scatter_kernel<4><<<blocks((long long)E * (FOUT / 4), B), B, 0, stream>>>(
      (const float4*)h2, src, dst, dinv, agg2, (long long)E * (FOUT / 4));
  finalize_logsoftmax_kernel<<<blocks(N, B), B, 0, stream>>>(
      (const float4*)agg2, (const float4*)h2, dinv, (const float4*)b2, (float4*)d_out, N);
}